// multi_gcn_time_51479478010106
// MI455X (gfx1250) — compile-verified
//
#include <hip/hip_runtime.h>

typedef __attribute__((ext_vector_type(2))) float v2f;
typedef __attribute__((ext_vector_type(8))) float v8f;
typedef __attribute__((ext_vector_type(4))) unsigned int v4u;
typedef __attribute__((ext_vector_type(8))) int v8i;
typedef __attribute__((ext_vector_type(4))) int v4i;

#define GV 512          // graph nodes
#define GL 13           // time length
#define GN 64           // batch
#define GC 32           // channels per block
#define VL (GV*GL)      // 6656
#define XNSTR (GC*VL)       // 212992  (x: per-n stride, 32 ch)
#define HNSTR (6*GC*VL)     // 1277952 (hop buf: per-n stride, 192 ch)
#define HZSTR (2*GC*VL)     // 425984  (per-support: 2 hop blocks)
#define LO 11           // conv output length
#define KTOT 672        // 224*3
#define CONVN (GN*GV*LO) // 360448

// ---------------------------------------------------------------------------
// TDM: 2D tile load Global->LDS. data_size=4B. Row padding via pad_interval /
// pad_amount codes (interval code c -> 2^(c+1) dwords, amount code a -> a+1
// dwords) so the LDS image directly matches a padded [rows][tile_w+pad] array.
// D# layout per CDNA5 ISA 8.3/8.4 (group0 128b, group1 256b, groups 2/3
// minimal). This toolchain exposes the 6-arg builtin; the 5th (int32x8) group
// is unused for a plain 2D load and passed as zeros.
// ---------------------------------------------------------------------------
__device__ __forceinline__ void tdm_load_2d_f32(unsigned lds_off, const float* gptr,
                                                unsigned tensor_w, unsigned tensor_h,
                                                unsigned tile_w, unsigned tile_h,
                                                unsigned row_stride,
                                                unsigned pad_int_code, unsigned pad_amt_code)
{
    unsigned long long ga = (unsigned long long)(size_t)gptr;
    v4u g0;
    g0[0] = 1u;                                        // count=1, no gather
    g0[1] = lds_off;                                   // lds_addr [63:32]
    g0[2] = (unsigned)(ga & 0xffffffffu);              // global_addr low
    g0[3] = (unsigned)((ga >> 32) & 0x1ffffffu)        // global_addr high (bits 120:96)
          | (2u << 30);                                // type=2 ("image")
    v8i g1;
    g1[0] = (int)((2u << 16)                           // data_size = 4B
                | (1u << 20)                           // pad_enable
                | (pad_int_code << 22)
                | (pad_amt_code << 25));
    g1[1] = (int)((tensor_w & 0xffffu) << 16);                                  // dim0[15:0]
    g1[2] = (int)(((tensor_w >> 16) & 0xffffu) | ((tensor_h & 0xffffu) << 16)); // dim0 hi | dim1 lo
    g1[3] = (int)(((tensor_h >> 16) & 0xffffu) | ((tile_w & 0xffffu) << 16));   // dim1 hi | tile_dim0
    g1[4] = (int)(tile_h & 0xffffu);                   // tile_dim1 (tile_dim2 = 0)
    g1[5] = (int)row_stride;                           // tensor_dim0_stride low 32
    g1[6] = 0;
    g1[7] = 0;
    v4i g2 = {1, 0, 0, 0};                             // tensor_dim2=1, rest unused
    v4i g3 = {0, 0, 0, 0};
    v8i g4 = {0, 0, 0, 0, 0, 0, 0, 0};                 // unused group (6-arg form)
    __builtin_amdgcn_tensor_load_to_lds(g0, g1, g2, g3, g4, 0);
}

// ---------------------------------------------------------------------------
// Diffusion GEMM:  out[w, j] = sum_v A[w,v] * in[v, j],   j = (n,c,l)
// WG: 256 thr (8 waves). Tile M=128 x N=64, K-chunk 64.
// Waves 4(M) x 2(N); each wave 32x32 via 4 accumulators (4 wmma / 4 b64 loads).
// A tile staged by the Tensor Data Mover; X tile gathered + stored transposed.
// ---------------------------------------------------------------------------
__global__ __launch_bounds__(256)
void diff_gemm(const float* __restrict__ Aall,
               const float* __restrict__ in, float* __restrict__ out,
               int in_nstride, int in_coff0, int in_zstr,
               int out_nstride, int out_coff0, int out_zstr)
{
    __shared__ float As[128][68];   // row-major, stride 68 (68/4 odd -> conflict-free b64)
    __shared__ float Xs[64][68];    // TRANSPOSED: [col][k], same stride trick

    const int z = blockIdx.z;
    const float* A = Aall + (size_t)z * GV * GV;
    const int in_coff  = in_coff0  + z * in_zstr;
    const int out_coff = out_coff0 + z * out_zstr;

    const int tid = threadIdx.x;
    const int wb = blockIdx.y * 128;    // M (w) tile base
    const int jb = blockIdx.x * 64;     // N (col) tile base

    // X-tile loader: one col per thread, 4 groups x 16 v-rows
    const int xcol = tid & 63;
    const int vg   = tid >> 6;
    const int j  = jb + xcol;
    const int q  = j / GL;
    const int l  = j - q * GL;
    const long inbase = (long)(q >> 5) * in_nstride + in_coff + (q & 31) * VL + l;

    // WMMA fragment mapping
    const int lane = tid & 31;
    const int wave = tid >> 5;
    const int m0 = (wave & 3) * 32;
    const int n0 = (wave >> 2) * 32;
    const int tlo = lane & 15;
    const int hi  = lane >> 4;

    v8f acc[2][2];
#pragma unroll
    for (int a = 0; a < 2; ++a)
#pragma unroll
        for (int b = 0; b < 2; ++b)
            acc[a][b] = (v8f){0.f,0.f,0.f,0.f,0.f,0.f,0.f,0.f};

    const unsigned as_lds = (unsigned)(size_t)&As[0][0];

    for (int k0 = 0; k0 < GV; k0 += 64) {
        // TDM: stage A tile 128 rows x 64 cols, padded to stride 68
        if (wave == 0) {
            tdm_load_2d_f32(as_lds, A + (size_t)wb * GV + k0,
                            /*tensor_w=*/GV, /*tensor_h=*/GV,
                            /*tile_w=*/64, /*tile_h=*/128,
                            /*row_stride=*/GV,
                            /*pad interval 64 dw*/5, /*pad 4 dw*/3);
        }
        // stage X tile transposed (v-strided global gather, b128 LDS stores)
#pragma unroll
        for (int i = 0; i < 4; ++i) {
            const int v = vg * 16 + i * 4;
            float4 t;
            t.x = in[inbase + (long)(k0 + v    ) * GL];
            t.y = in[inbase + (long)(k0 + v + 1) * GL];
            t.z = in[inbase + (long)(k0 + v + 2) * GL];
            t.w = in[inbase + (long)(k0 + v + 3) * GL];
            *(float4*)&Xs[xcol][v] = t;
        }
        __builtin_amdgcn_s_wait_tensorcnt(0);
        __syncthreads();

#pragma unroll
        for (int kk = 0; kk < 64; kk += 4) {
            v2f a0, a1, b0, b1;
            a0 = *(const v2f*)&As[m0 + tlo     ][kk + 2*hi];
            a1 = *(const v2f*)&As[m0 + 16 + tlo][kk + 2*hi];
            b0 = *(const v2f*)&Xs[n0 + tlo     ][kk + 2*hi];
            b1 = *(const v2f*)&Xs[n0 + 16 + tlo][kk + 2*hi];
            acc[0][0] = __builtin_amdgcn_wmma_f32_16x16x4_f32(false, a0, false, b0,
                                                              (short)0, acc[0][0], false, false);
            acc[0][1] = __builtin_amdgcn_wmma_f32_16x16x4_f32(false, a0, false, b1,
                                                              (short)0, acc[0][1], false, false);
            acc[1][0] = __builtin_amdgcn_wmma_f32_16x16x4_f32(false, a1, false, b0,
                                                              (short)0, acc[1][0], false, false);
            acc[1][1] = __builtin_amdgcn_wmma_f32_16x16x4_f32(false, a1, false, b1,
                                                              (short)0, acc[1][1], false, false);
        }
        __syncthreads();
    }

    // write back: accumulator element r is (M = m0+mh*16+r+8*hi, N = n0+nh*16+tlo)
#pragma unroll
    for (int nh = 0; nh < 2; ++nh) {
        const int jj = jb + n0 + nh * 16 + tlo;
        const int q2 = jj / GL;
        const int l2 = jj - q2 * GL;
        const long ob = (long)(q2 >> 5) * out_nstride + out_coff + (q2 & 31) * VL + l2;
#pragma unroll
        for (int mh = 0; mh < 2; ++mh) {
#pragma unroll
            for (int r = 0; r < 8; ++r) {
                const int w = wb + m0 + mh * 16 + r + 8 * hi;
                out[ob + (long)w * GL] = acc[mh][nh][r];
            }
        }
    }
}

// ---------------------------------------------------------------------------
// Conv GEMM: y[o, j] = b[o] + sum_{k=(cc,kt)} W[o,k] * h[cc, base(j)+kt]
// h is virtual: cc<32 -> x, else hop buffer block cc-32.
// WG tile M=32 x N=128, K-chunk 48. 8 waves (2M x 4N), 2 accumulators each.
// ---------------------------------------------------------------------------
__global__ __launch_bounds__(256)
void conv_gemm(const float* __restrict__ x, const float* __restrict__ hb,
               const float* __restrict__ W, const float* __restrict__ bias,
               float* __restrict__ y)
{
    __shared__ float Ws[32][52];    // row-major, stride 52 (52/4 odd -> conflict-free)
    __shared__ float Xs[128][52];   // TRANSPOSED: [col][k]

    const int tid = threadIdx.x;
    const int jb = blockIdx.x * 128;

    // X loader: col per thread, 2 groups x 24 k-rows
    const int xcol = tid & 127;
    const int kg   = tid >> 7;
    const int j   = jb + xcol;
    const int n   = j / (GV * LO);
    const int rem = j - n * (GV * LO);
    const int v   = rem / LO;
    const int lo  = rem - v * LO;
    const int base = v * GL + lo;
    const float* xn = x  + (long)n * XNSTR;
    const float* hn = hb + (long)n * HNSTR;

    const int lane = tid & 31;
    const int wave = tid >> 5;
    const int m0 = (wave & 1) * 16;
    const int n0 = (wave >> 1) * 32;
    const int tlo = lane & 15;
    const int hi  = lane >> 4;

    v8f acc0 = {0.f,0.f,0.f,0.f,0.f,0.f,0.f,0.f};
    v8f acc1 = {0.f,0.f,0.f,0.f,0.f,0.f,0.f,0.f};

    for (int k0 = 0; k0 < KTOT; k0 += 48) {
        // stage W chunk 32x48 (1536 elems, 6 per thread)
#pragma unroll
        for (int i = 0; i < 6; ++i) {
            const int e = tid + i * 256;
            const int o = e / 48;
            const int kk = e - o * 48;
            Ws[o][kk] = W[o * KTOT + k0 + kk];
        }
        // stage X chunk transposed, 4 k at a time
#pragma unroll
        for (int i = 0; i < 6; ++i) {
            const int kbase = kg * 24 + i * 4;
            float4 t;
#pragma unroll
            for (int d = 0; d < 4; ++d) {
                const int kglob = k0 + kbase + d;      // k0 % 3 == 0
                const int cc = kglob / 3;
                const int kt = kglob - cc * 3;
                const float val = (cc < GC) ? xn[cc * VL + base + kt]
                                            : hn[(cc - GC) * VL + base + kt];
                ((float*)&t)[d] = val;
            }
            *(float4*)&Xs[xcol][kbase] = t;
        }
        __syncthreads();

#pragma unroll
        for (int kk = 0; kk < 48; kk += 4) {
            v2f a, b0, b1;
            a  = *(const v2f*)&Ws[m0 + tlo     ][kk + 2*hi];
            b0 = *(const v2f*)&Xs[n0 + tlo     ][kk + 2*hi];
            b1 = *(const v2f*)&Xs[n0 + 16 + tlo][kk + 2*hi];
            acc0 = __builtin_amdgcn_wmma_f32_16x16x4_f32(false, a, false, b0,
                                                         (short)0, acc0, false, false);
            acc1 = __builtin_amdgcn_wmma_f32_16x16x4_f32(false, a, false, b1,
                                                         (short)0, acc1, false, false);
        }
        __syncthreads();
    }

#pragma unroll
    for (int half = 0; half < 2; ++half) {
        const int jj = jb + n0 + half * 16 + tlo;
        const int n2 = jj / (GV * LO);
        const int rem2 = jj - n2 * (GV * LO);
        const int v2 = rem2 / LO;
        const int lo2 = rem2 - v2 * LO;
        const v8f acc = half ? acc1 : acc0;
#pragma unroll
        for (int r = 0; r < 8; ++r) {
            const int o = m0 + r + 8 * hi;
            y[(long)n2 * (GC * GV * LO) + o * (GV * LO) + v2 * LO + lo2] = acc[r] + bias[o];
        }
    }
}

// ---------------------------------------------------------------------------
extern "C" void kernel_launch(void* const* d_in, const int* in_sizes, int n_in,
                              void* d_out, int out_size, void* d_ws, size_t ws_size,
                              hipStream_t stream)
{
    (void)in_sizes; (void)n_in; (void)out_size; (void)ws_size;
    const float* x       = (const float*)d_in[0];
    const float* support = (const float*)d_in[1];
    const float* W       = (const float*)d_in[2];
    const float* b       = (const float*)d_in[3];
    float* y  = (float*)d_out;
    float* hb = (float*)d_ws;   // [64][192][512][13] hop blocks: A0x,A0^2x,A1x,A1^2x,A2x,A2^2x

    const dim3 dgrid(26624 / 64, GV / 128, 3);   // (416, 4, 3)

    // hop 1: in = x (32-ch layout), out = hop block 2z
    diff_gemm<<<dgrid, 256, 0, stream>>>(support, x, hb,
                                         XNSTR, 0, 0,
                                         HNSTR, 0, HZSTR);
    // hop 2: in = hop block 2z, out = hop block 2z+1
    diff_gemm<<<dgrid, 256, 0, stream>>>(support, hb, hb,
                                         HNSTR, 0, HZSTR,
                                         HNSTR, GC * VL, HZSTR);
    // temporal conv over virtual h = [x | hop blocks]
    conv_gemm<<<dim3(CONVN / 128, 1, 1), 256, 0, stream>>>(x, hb, W, b, y);
}